// CRPS_88493506167040
// MI455X (gfx1250) — compile-verified
//
#include <hip/hip_runtime.h>

typedef __attribute__((ext_vector_type(2))) float v2f;
typedef __attribute__((ext_vector_type(4))) float v4f;
typedef __attribute__((ext_vector_type(8))) float v8f;

#define NB    2
#define ENS   16
#define NT    16
#define NLAT  128
#define NLON  256
#define IMG   (NLAT * NLON)            // 32768 points per (b,t) image
#define NBT   (NB * NT)                // 32 output values
#define NBLK  8                        // partial blocks per (b,t): K=8 -> 2 chained 16x16x4 WMMAs
#define TPB   256
#define PTS_PER_BLOCK (IMG / NBLK)     // 4096
#define F4_ITERS (PTS_PER_BLOCK / (TPB * 4))  // 4 float4 iterations per thread

__device__ __forceinline__ void ce(float& a, float& b) {
    float lo = fminf(a, b);
    float hi = fmaxf(a, b);
    a = lo; b = hi;
}

// Batcher odd-even mergesort, 16 elements, 63 compare-exchanges (verified by
// recursive construction: 4x sort4 + 2x merge(4,4) + merge(8,8)).
__device__ __forceinline__ void sort16(float x[ENS]) {
    // sort4 on each quartet (layers 1-3)
    ce(x[0],x[1]); ce(x[2],x[3]); ce(x[4],x[5]); ce(x[6],x[7]);
    ce(x[8],x[9]); ce(x[10],x[11]); ce(x[12],x[13]); ce(x[14],x[15]);
    ce(x[0],x[2]); ce(x[1],x[3]); ce(x[4],x[6]); ce(x[5],x[7]);
    ce(x[8],x[10]); ce(x[9],x[11]); ce(x[12],x[14]); ce(x[13],x[15]);
    ce(x[1],x[2]); ce(x[5],x[6]); ce(x[9],x[10]); ce(x[13],x[14]);
    // merge(4,4) on [0..7] and [8..15] (layers 4-6)
    ce(x[0],x[4]); ce(x[2],x[6]); ce(x[1],x[5]); ce(x[3],x[7]);
    ce(x[8],x[12]); ce(x[10],x[14]); ce(x[9],x[13]); ce(x[11],x[15]);
    ce(x[2],x[4]); ce(x[3],x[5]); ce(x[10],x[12]); ce(x[11],x[13]);
    ce(x[1],x[2]); ce(x[3],x[4]); ce(x[5],x[6]);
    ce(x[9],x[10]); ce(x[11],x[12]); ce(x[13],x[14]);
    // merge(8,8) (layers 7-10)
    ce(x[0],x[8]); ce(x[1],x[9]); ce(x[2],x[10]); ce(x[3],x[11]);
    ce(x[4],x[12]); ce(x[5],x[13]); ce(x[6],x[14]); ce(x[7],x[15]);
    ce(x[4],x[8]); ce(x[5],x[9]); ce(x[6],x[10]); ce(x[7],x[11]);
    ce(x[2],x[4]); ce(x[3],x[5]); ce(x[6],x[8]); ce(x[7],x[9]);
    ce(x[10],x[12]); ce(x[11],x[13]);
    ce(x[1],x[2]); ce(x[3],x[4]); ce(x[5],x[6]); ce(x[7],x[8]);
    ce(x[9],x[10]); ce(x[11],x[12]); ce(x[13],x[14]);
}

// Stage 1: streaming kernel. grid = (NBLK, NBT). Each thread handles
// F4_ITERS groups of 4 consecutive-lon points (TPB*4 apart); all 16 ensemble values live in regs.
// Writes one deterministic partial sum per block to part[bt*NBLK + blk].
__global__ __launch_bounds__(TPB) void crps_partial(
    const float* __restrict__ pred, const float* __restrict__ tgt,
    const float* __restrict__ latw, float* __restrict__ part)
{
    const int blk = blockIdx.x;          // 0..NBLK-1
    const int bt  = blockIdx.y;          // b*NT + t
    const int b   = bt >> 4;
    const int t   = bt & 15;

    const float* timg  = tgt + (size_t)bt * IMG;
    const float* pbase = pred + ((size_t)b * ENS * NT + (size_t)t) * IMG;
    const size_t estride = (size_t)NT * IMG;   // ensemble stride: 2 MiB

    float acc = 0.0f;

#pragma unroll
    for (int it = 0; it < F4_ITERS; ++it) {
        const int p0 = blk * PTS_PER_BLOCK + it * (TPB * 4) + (int)threadIdx.x * 4;
        const float lw = latw[p0 >> 8];  // p0 % 4 == 0, never crosses a lat row

        const v4f tv = *reinterpret_cast<const v4f*>(timg + p0);

        v4f pv[ENS];
#pragma unroll
        for (int e = 0; e < ENS; ++e)
            pv[e] = __builtin_nontemporal_load(
                reinterpret_cast<const v4f*>(pbase + (size_t)e * estride + p0));

#pragma unroll
        for (int j = 0; j < 4; ++j) {
            float x[ENS];
#pragma unroll
            for (int e = 0; e < ENS; ++e) x[e] = pv[e][j];
            const float tj = tv[j];

            float skill = 0.0f;
#pragma unroll
            for (int e = 0; e < ENS; ++e) skill += fabsf(x[e] - tj);

            sort16(x);

            float s = 0.0f;
#pragma unroll
            for (int e = 0; e < ENS; ++e) s = fmaf(2.0f * e - 15.0f, x[e], s);

            // crps point value = lw * (skill/16 - 0.5 * (2*s / (16*15)))
            acc = fmaf(lw, skill * (1.0f / 16.0f) - s * (1.0f / 240.0f), acc);
        }
    }

    // Deterministic reduction: wave32 shuffle tree, then fixed-order LDS sum.
#pragma unroll
    for (int off = 16; off > 0; off >>= 1)
        acc += __shfl_xor(acc, off, 32);

    __shared__ float red[TPB / 32];
    const int lane = threadIdx.x & 31;
    const int wid  = threadIdx.x >> 5;
    if (lane == 0) red[wid] = acc;
    __syncthreads();
    if (threadIdx.x == 0) {
        float s = 0.0f;
#pragma unroll
        for (int w = 0; w < TPB / 32; ++w) s += red[w];
        part[bt * NBLK + blk] = s;
    }
}

// Stage 2: single wave32. Reduce the 32x8 partial matrix with chained
// V_WMMA_F32_16X16X4_F32 row-sums (A = 16x4 tile of partials, B = ones),
// then cumulative-average over t.
__global__ __launch_bounds__(32) void crps_finalize(
    const float* __restrict__ part, float* __restrict__ out)
{
    __shared__ float sums[NBT];
    const int l  = (int)threadIdx.x;
    const int m  = l & 15;   // matrix row within half
    const int hi = l >> 4;   // lane group selects K half per documented A layout

    v2f ones; ones.x = 1.0f; ones.y = 1.0f;

#pragma unroll
    for (int h = 0; h < 2; ++h) {              // rows 0-15, then 16-31
        const float* base = part + h * 16 * NBLK;
        // f32 16x4 A layout: VGPR0 = K0 (lanes 0-15) / K2 (lanes 16-31),
        //                    VGPR1 = K1 (lanes 0-15) / K3 (lanes 16-31)
        v2f a0, a1;
        a0.x = base[m * NBLK + (hi ? 2 : 0)];
        a0.y = base[m * NBLK + (hi ? 3 : 1)];
        a1.x = base[m * NBLK + 4 + (hi ? 2 : 0)];
        a1.y = base[m * NBLK + 4 + (hi ? 3 : 1)];

        v8f c = {};
        c = __builtin_amdgcn_wmma_f32_16x16x4_f32(false, a0, false, ones,
                                                  (short)0, c, false, false);
        c = __builtin_amdgcn_wmma_f32_16x16x4_f32(false, a1, false, ones,
                                                  (short)0, c, false, false);

        // D layout: lanes 0-15 VGPR r = row r (N=lane); lanes 16-31 = row 8+r.
        if (m == 0) {
#pragma unroll
            for (int r = 0; r < 8; ++r)
                sums[h * 16 + hi * 8 + r] = c[r];
        }
    }
    __syncthreads();

    if (l == 0) {
        for (int b = 0; b < NB; ++b) {
            float run = 0.0f;
            for (int t = 0; t < NT; ++t) {
                run += sums[b * NT + t] * (1.0f / (float)IMG);  // spatial mean
                out[b * NT + t] = run / (float)(t + 1);         // cumulative avg
            }
        }
    }
}

extern "C" void kernel_launch(void* const* d_in, const int* in_sizes, int n_in,
                              void* d_out, int out_size, void* d_ws, size_t ws_size,
                              hipStream_t stream) {
    (void)in_sizes; (void)n_in; (void)out_size; (void)ws_size;
    const float* pred = (const float*)d_in[0];
    const float* tgt  = (const float*)d_in[1];
    const float* latw = (const float*)d_in[2];
    float* out  = (float*)d_out;
    float* part = (float*)d_ws;   // NBT * NBLK = 256 floats

    dim3 grid(NBLK, NBT);
    crps_partial<<<grid, TPB, 0, stream>>>(pred, tgt, latw, part);
    crps_finalize<<<1, 32, 0, stream>>>(part, out);
}